// OpGrad_67783173865700
// MI455X (gfx1250) — compile-verified
//
#include <hip/hip_runtime.h>
#include <math.h>

#ifndef __has_builtin
#define __has_builtin(x) 0
#endif

#if defined(__HIP_DEVICE_COMPILE__) && __has_builtin(__builtin_amdgcn_global_load_async_to_lds_b128)
#define HAVE_ASYNC_LDS 1
#else
#define HAVE_ASYNC_LDS 0
#endif

#define KEPS 1e-6f
#define MAX_CELL_FLOATS 2304 /* 256 cells * 9 floats = 9216 B, fits easily in 320KB LDS */
#define BLOCK 256

typedef int v4i __attribute__((ext_vector_type(4)));
typedef __attribute__((address_space(1))) v4i* gv4i_p;  // global
typedef __attribute__((address_space(3))) v4i* lv4i_p;  // LDS

struct F3 { float x, y, z; };

__device__ __forceinline__ F3 make_f3(float a, float b, float c) { F3 r{a,b,c}; return r; }

__device__ __forceinline__ F3 cart(const float* __restrict__ c, F3 v) {
  // e_j = sum_i v_i * cell[i][j], cell row-major (3x3)
  F3 e;
  e.x = v.x * c[0] + v.y * c[3] + v.z * c[6];
  e.y = v.x * c[1] + v.y * c[4] + v.z * c[7];
  e.z = v.x * c[2] + v.y * c[5] + v.z * c[8];
  return e;
}

__device__ __forceinline__ float dot3(F3 a, F3 b) { return a.x*b.x + a.y*b.y + a.z*b.z; }

__device__ __forceinline__ F3 cross3(F3 a, F3 b) {
  return make_f3(a.y*b.z - a.z*b.y, a.z*b.x - a.x*b.z, a.x*b.y - a.y*b.x);
}

// Cooperative stage of the whole cell table into LDS via the CDNA5 async
// global->LDS path (ASYNCcnt-tracked), with portable fallbacks.
__device__ __forceinline__ void stage_cells(const float* __restrict__ cell,
                                            float* __restrict__ s_cell,
                                            int ncellf) {
  const int nvec = ncellf >> 2;          // 16-byte chunks
  for (int v = threadIdx.x; v < nvec; v += BLOCK) {
#if HAVE_ASYNC_LDS
    __builtin_amdgcn_global_load_async_to_lds_b128(
        (gv4i_p)(cell + 4 * v),
        (lv4i_p)(s_cell + 4 * v),
        /*offset=*/0, /*cpol=*/0);
#else
    ((float4*)s_cell)[v] = ((const float4*)cell)[v];
#endif
  }
  for (int k = (nvec << 2) + threadIdx.x; k < ncellf; k += BLOCK)
    s_cell[k] = cell[k];
#if HAVE_ASYNC_LDS
#if __has_builtin(__builtin_amdgcn_s_wait_asynccnt)
  __builtin_amdgcn_s_wait_asynccnt(0);
#else
  asm volatile("s_wait_asynccnt 0" ::: "memory");
#endif
#endif
  __syncthreads();
}

// ---------------- edges: grad_distance -> (E,1,3,3) ----------------
__global__ void __launch_bounds__(BLOCK)
edges_kernel(const float* __restrict__ cell, int ncellf,
             const int* __restrict__ batch,
             const float* __restrict__ xin,
             float* __restrict__ out, int E) {
  __shared__ __align__(16) float s_cell[MAX_CELL_FLOATS];
  const bool use_lds = (ncellf <= MAX_CELL_FLOATS);
  if (use_lds) stage_cells(cell, s_cell, ncellf);

  const int i = blockIdx.x * BLOCK + threadIdx.x;
  if (i >= E) return;

  const int b = batch[i];
  const float* __restrict__ c = use_lds ? (s_cell + 9 * b) : (cell + 9 * b);

  const size_t i3 = (size_t)i * 3;
  F3 x = make_f3(xin[i3], xin[i3 + 1], xin[i3 + 2]);
  F3 e = cart(c, x);
  const float inv = 1.0f / sqrtf(dot3(e, e));
  F3 u = make_f3(e.x * inv, e.y * inv, e.z * inv);

  float* __restrict__ o = out + (size_t)i * 9;
  o[0] = x.x * u.x; o[1] = x.x * u.y; o[2] = x.x * u.z;
  o[3] = x.y * u.x; o[4] = x.y * u.y; o[5] = x.y * u.z;
  o[6] = x.z * u.x; o[7] = x.z * u.y; o[8] = x.z * u.z;
}

// ------------- triplets: grad_angle + grad_area -> (T,2,3,3) -------------
__global__ void __launch_bounds__(BLOCK)
triplets_kernel(const float* __restrict__ cell, int ncellf,
                const int* __restrict__ batch,
                const float* __restrict__ xin,
                const float* __restrict__ yin,
                float* __restrict__ out, int T) {
  __shared__ __align__(16) float s_cell[MAX_CELL_FLOATS];
  const bool use_lds = (ncellf <= MAX_CELL_FLOATS);
  if (use_lds) stage_cells(cell, s_cell, ncellf);

  const int i = blockIdx.x * BLOCK + threadIdx.x;
  if (i >= T) return;

  const int b = batch[i];
  const float* __restrict__ c = use_lds ? (s_cell + 9 * b) : (cell + 9 * b);

  const size_t i3 = (size_t)i * 3;
  F3 x = make_f3(xin[i3], xin[i3 + 1], xin[i3 + 2]);
  F3 y = make_f3(yin[i3], yin[i3 + 1], yin[i3 + 2]);

  F3 e = cart(c, x);
  F3 f = cart(c, y);

  // ---- grad_angle ----
  const float de  = sqrtf(dot3(e, e));
  const float df  = sqrtf(dot3(f, f));
  const float ide = 1.0f / de;
  const float idf = 1.0f / df;
  F3 ue = make_f3(e.x * ide, e.y * ide, e.z * ide);
  F3 uf = make_f3(f.x * idf, f.y * idf, f.z * idf);
  float cs = dot3(ue, uf);
  cs = fminf(fmaxf(cs, -1.0f + KEPS), 1.0f - KEPS);
  const float coef = -1.0f / sqrtf(1.0f - cs * cs);
  // dth_de = coef*(uf - cs*ue)/de ; dth_df = coef*(ue - cs*uf)/df
  F3 a = make_f3(coef * (uf.x - cs * ue.x) * ide,
                 coef * (uf.y - cs * ue.y) * ide,
                 coef * (uf.z - cs * ue.z) * ide);
  F3 g = make_f3(coef * (ue.x - cs * uf.x) * idf,
                 coef * (ue.y - cs * uf.y) * idf,
                 coef * (ue.z - cs * uf.z) * idf);

  // ---- grad_area ----
  F3 n = cross3(e, f);
  const float A    = sqrtf(dot3(n, n));
  const float invA = 1.0f / fmaxf(A, KEPS);
  F3 nh = make_f3(n.x * invA, n.y * invA, n.z * invA);
  F3 p = cross3(f, nh);   // dA/de
  F3 q = cross3(nh, e);   // dA/df

  float* __restrict__ o = out + (size_t)i * 18;
  // op 0: angle = outer(x, a) + outer(y, g)
  o[0]  = x.x * a.x + y.x * g.x; o[1]  = x.x * a.y + y.x * g.y; o[2]  = x.x * a.z + y.x * g.z;
  o[3]  = x.y * a.x + y.y * g.x; o[4]  = x.y * a.y + y.y * g.y; o[5]  = x.y * a.z + y.y * g.z;
  o[6]  = x.z * a.x + y.z * g.x; o[7]  = x.z * a.y + y.z * g.y; o[8]  = x.z * a.z + y.z * g.z;
  // op 1: area = outer(x, p) + outer(y, q)
  o[9]  = x.x * p.x + y.x * q.x; o[10] = x.x * p.y + y.x * q.y; o[11] = x.x * p.z + y.x * q.z;
  o[12] = x.y * p.x + y.y * q.x; o[13] = x.y * p.y + y.y * q.y; o[14] = x.y * p.z + y.y * q.z;
  o[15] = x.z * p.x + y.z * q.x; o[16] = x.z * p.y + y.z * q.y; o[17] = x.z * p.z + y.z * q.z;
}

extern "C" void kernel_launch(void* const* d_in, const int* in_sizes, int n_in,
                              void* d_out, int out_size, void* d_ws, size_t ws_size,
                              hipStream_t stream) {
  (void)n_in; (void)out_size; (void)d_ws; (void)ws_size;
  const float* cell  = (const float*)d_in[0];  // (B,3,3)
  const int*   be    = (const int*)  d_in[1];  // (E,)
  const float* exij  = (const float*)d_in[2];  // (E,3)
  const int*   bt    = (const int*)  d_in[3];  // (T,)
  const float* txij  = (const float*)d_in[4];  // (T,3)
  const float* txik  = (const float*)d_in[5];  // (T,3)

  const int ncellf = in_sizes[0];
  const int E      = in_sizes[1];
  const int T      = in_sizes[3];

  float* out_edges = (float*)d_out;                       // E*9 floats
  float* out_trip  = out_edges + (size_t)E * 9;           // T*18 floats

  if (E > 0) {
    dim3 grid((E + BLOCK - 1) / BLOCK);
    edges_kernel<<<grid, BLOCK, 0, stream>>>(cell, ncellf, be, exij, out_edges, E);
  }
  if (T > 0) {
    dim3 grid((T + BLOCK - 1) / BLOCK);
    triplets_kernel<<<grid, BLOCK, 0, stream>>>(cell, ncellf, bt, txij, txik, out_trip, T);
  }
}